// Encoder_1108101562862
// MI455X (gfx1250) — compile-verified
//
#include <hip/hip_runtime.h>
#include <cstdint>

// Problem dims
#define B_   64
#define I_   256
#define H_   512
#define T_   512
#define NBLK 32      // persistent workgroups; one per WGP (212KB LDS each)
#define JW   16      // hidden units per workgroup (H_/NBLK)

typedef __attribute__((ext_vector_type(2))) float v2f;
typedef __attribute__((ext_vector_type(8))) float v8f;

// ---------------- workspace layout ----------------
// [0, 33554432)                 : xT  (T, B, I) f32
// [33554432, 33816576)          : hbuf, double buffer 2 x (B, H) f32
// [33816576, +4)                : global barrier ticket counter
#define WS_XT    0
#define WS_HBUF  33554432
#define WS_BAR   (33554432 + 262144)

__global__ void init_ws_kernel(unsigned* __restrict__ bar) {
    if (threadIdx.x == 0) *bar = 0u;
}

// Tiled transpose: x (B, I, T) -> xT (T, B, I), coalesced both sides via LDS.
__global__ __launch_bounds__(256) void transpose_x_kernel(
    const float* __restrict__ x, float* __restrict__ xT)
{
    __shared__ float tile[32][33];
    const int b  = blockIdx.z;
    const int i0 = blockIdx.y * 32;
    const int t0 = blockIdx.x * 32;
    const int tx = threadIdx.x;      // 32
    const int ty = threadIdx.y;      // 8
    const float* src = x + (size_t)b * I_ * T_;
#pragma unroll
    for (int r = 0; r < 4; ++r)
        tile[ty + 8 * r][tx] = src[(size_t)(i0 + ty + 8 * r) * T_ + t0 + tx];
    __syncthreads();
#pragma unroll
    for (int r = 0; r < 4; ++r)
        xT[(size_t)(t0 + ty + 8 * r) * (B_ * I_) + (size_t)b * I_ + i0 + tx] =
            tile[tx][ty + 8 * r];
}

__device__ __forceinline__ float sigm(float v) { return 1.0f / (1.0f + __expf(-v)); }

// Persistent LSTM scan. 32 blocks x 256 threads (8 waves). Block wg owns hidden
// units [wg*16, wg*16+16) i.e. 64 gate rows {g*H + J0 + j : g in 0..3, j in 0..15}.
// Wave layout: mb = w&3 (batch block of 16), gpair = w>>2 -> gate groups
// {2*gpair, 2*gpair+1}. Two f32 16x16 accumulators per wave.
__global__ __launch_bounds__(256) void lstm_scan_kernel(
    const float* __restrict__ xT,   const float* __restrict__ W_ih,
    const float* __restrict__ W_hh, const float* __restrict__ b_ih,
    const float* __restrict__ b_hh, const float* __restrict__ h0,
    const float* __restrict__ c0,   float* __restrict__ hbuf,
    unsigned* __restrict__ bar,     float* __restrict__ out_last,
    float* __restrict__ out_enc)
{
    extern __shared__ char smem[];
    // Pair-transposed weight slices: Wp[pair][r] = (W[row(r)][2p], W[row(r)][2p+1])
    v2f*   Whh_p = (v2f*)smem;                              // 256*64 v2f = 131072 B
    v2f*   Wih_p = (v2f*)(smem + 131072);                   // 128*64 v2f =  65536 B
    float* gates = (float*)(smem + 131072 + 65536);         // 64*64 f32  =  16384 B
    float* c_lds = (float*)(smem + 131072 + 65536 + 16384); // 1024 f32   =   4096 B

    const int tid = threadIdx.x;
    const int J0  = blockIdx.x * JW;

    // ---- load weight slices into LDS (once for all 512 steps) ----
    for (int idx = tid; idx < 64 * 256; idx += 256) {
        const int r = idx >> 8, p = idx & 255;          // local row, pair index
        const int grp = r >> 4, rr = r & 15;
        const float* row = W_hh + (size_t)(grp * H_ + J0 + rr) * H_;
        Whh_p[p * 64 + r] = *(const v2f*)(row + 2 * p);
    }
    for (int idx = tid; idx < 64 * 128; idx += 256) {
        const int r = idx >> 7, p = idx & 127;
        const int grp = r >> 4, rr = r & 15;
        const float* row = W_ih + (size_t)(grp * H_ + J0 + rr) * I_;
        Wih_p[p * 64 + r] = *(const v2f*)(row + 2 * p);
    }
    for (int e = tid; e < 1024; e += 256) {
        const int b = e >> 4, j = e & 15;
        c_lds[e] = c0[(size_t)b * H_ + J0 + j];
    }
    __syncthreads();

    // ---- WMMA fragment coordinates (CDNA5 ISA 7.12.2, wave32) ----
    const int lane = tid & 31;
    const int w    = tid >> 5;
    const int hl   = lane >> 4;        // half: K/M offset selector
    const int n    = lane & 15;        // N (gate col) for B/C/D, M row for A
    const int mb   = w & 3;            // batch block
    const int g0   = (w >> 2) * 2;     // gate groups
    const int g1   = g0 + 1;

    const float bias0 = b_ih[g0 * H_ + J0 + n] + b_hh[g0 * H_ + J0 + n];
    const float bias1 = b_ih[g1 * H_ + J0 + n] + b_hh[g1 * H_ + J0 + n];

    const int arow = mb * 16 + n;      // A-matrix row (batch index)

    for (int t = 0; t < T_; ++t) {
        const float* hsrc = (t == 0) ? h0 : (hbuf + (size_t)((t - 1) & 1) * (B_ * H_));
        v8f acc0 = {}; v8f acc1 = {};

        // gates += h @ W_hh^T (K = 512)
        const float* ha = hsrc + (size_t)arow * H_ + 2 * hl;
        for (int k0 = 0; k0 < H_; k0 += 4) {
            const v2f a  = *(const v2f*)(ha + k0);
            const int p  = (k0 >> 1) + hl;
            const v2f bA = Whh_p[p * 64 + g0 * 16 + n];
            const v2f bB = Whh_p[p * 64 + g1 * 16 + n];
            acc0 = __builtin_amdgcn_wmma_f32_16x16x4_f32(false, a, false, bA,
                                                         (short)0, acc0, false, false);
            acc1 = __builtin_amdgcn_wmma_f32_16x16x4_f32(false, a, false, bB,
                                                         (short)0, acc1, false, false);
        }
        // gates += x_t @ W_ih^T (K = 256); prefetch next step's x rows
        const float* xa = xT + (size_t)t * (B_ * I_) + (size_t)arow * I_ + 2 * hl;
        if (t + 1 < T_)
            __builtin_prefetch(xT + (size_t)(t + 1) * (B_ * I_) + (size_t)arow * I_, 0, 0);
        for (int k0 = 0; k0 < I_; k0 += 4) {
            const v2f a  = *(const v2f*)(xa + k0);
            const int p  = (k0 >> 1) + hl;
            const v2f bA = Wih_p[p * 64 + g0 * 16 + n];
            const v2f bB = Wih_p[p * 64 + g1 * 16 + n];
            acc0 = __builtin_amdgcn_wmma_f32_16x16x4_f32(false, a, false, bA,
                                                         (short)0, acc0, false, false);
            acc1 = __builtin_amdgcn_wmma_f32_16x16x4_f32(false, a, false, bB,
                                                         (short)0, acc1, false, false);
        }

        // spill D-tiles (+bias) to LDS: VGPR r -> M = r + 8*hl, N = n
#pragma unroll
        for (int r = 0; r < 8; ++r) {
            const int m = mb * 16 + r + hl * 8;
            gates[m * 64 + g0 * 16 + n] = acc0[r] + bias0;
            gates[m * 64 + g1 * 16 + n] = acc1[r] + bias1;
        }
        __syncthreads();

        // elementwise cell update (PyTorch gate order i, f, g, o)
        float* hdst = hbuf + (size_t)(t & 1) * (B_ * H_);
        for (int e = tid; e < 1024; e += 256) {
            const int b = e >> 4, j = e & 15;
            const float ig = sigm(gates[b * 64 +      j]);
            const float fg = sigm(gates[b * 64 + 16 + j]);
            const float gg = tanhf(gates[b * 64 + 32 + j]);
            const float og = sigm(gates[b * 64 + 48 + j]);
            const float cn = fg * c_lds[e] + ig * gg;
            const float hn = og * tanhf(cn);
            c_lds[e] = cn;
            hdst[(size_t)b * H_ + J0 + j] = hn;
            out_enc[((size_t)b * H_ + (J0 + j)) * T_ + t] = hn;
            if (t == T_ - 1) out_last[(size_t)b * H_ + J0 + j] = hn;
        }
        __syncthreads();

        // grid-wide barrier: monotonic ticket (no reset race, reusable)
        if (tid == 0) {
            __threadfence();
            const unsigned arrive = atomicAdd(bar, 1u) + 1u;
            const unsigned target = ((arrive + NBLK - 1u) / NBLK) * NBLK;
            while (atomicAdd(bar, 0u) < target) { __builtin_amdgcn_s_sleep(1); }
            __threadfence();
        }
        __syncthreads();
    }
}

extern "C" void kernel_launch(void* const* d_in, const int* in_sizes, int n_in,
                              void* d_out, int out_size, void* d_ws, size_t ws_size,
                              hipStream_t stream) {
    (void)in_sizes; (void)n_in; (void)out_size; (void)ws_size;
    const float* x    = (const float*)d_in[0];
    const float* W_ih = (const float*)d_in[1];
    const float* W_hh = (const float*)d_in[2];
    const float* b_ih = (const float*)d_in[3];
    const float* b_hh = (const float*)d_in[4];
    const float* h0   = (const float*)d_in[5];
    const float* c0   = (const float*)d_in[6];
    float* out = (float*)d_out;

    char* ws = (char*)d_ws;
    float*    xT   = (float*)(ws + WS_XT);
    float*    hbuf = (float*)(ws + WS_HBUF);
    unsigned* bar  = (unsigned*)(ws + WS_BAR);

    init_ws_kernel<<<1, 32, 0, stream>>>(bar);
    transpose_x_kernel<<<dim3(T_ / 32, I_ / 32, B_), dim3(32, 8), 0, stream>>>(x, xT);

    const size_t smem = 131072 + 65536 + 16384 + 4096;   // 217088 B < 320 KB WGP LDS
    (void)hipFuncSetAttribute(reinterpret_cast<const void*>(lstm_scan_kernel),
                              hipFuncAttributeMaxDynamicSharedMemorySize, (int)smem);
    lstm_scan_kernel<<<NBLK, 256, smem, stream>>>(
        xT, W_ih, W_hh, b_ih, b_hh, h0, c0, hbuf, bar,
        /*out_last=*/out, /*out_enc=*/out + (size_t)B_ * H_);
}